// DTWShpTime_46059229282878
// MI455X (gfx1250) — compile-verified
//
#include <hip/hip_runtime.h>

typedef __attribute__((ext_vector_type(2))) float v2f;
typedef __attribute__((ext_vector_type(8))) float v8f;

#define NN   160
#define BC   512
#define CCH  8
#define BB   64
#define GAMMA_F 0.01f
#define INV_G   100.0f
#define BIGF    1e10f

// One workgroup (160 threads = 5 wave32) per (b,c) problem.
// Phase 1: stage t (target) and p (input) series in LDS.
// Phase 2: build D[i][j] = (t_i - p_j)^2 as a K=4 GEMM via v_wmma_f32_16x16x4_f32:
//          A row m = [t_m^2, t_m, 1, 0], B col n = [1, -2 p_n, p_n^2, 0].
//          Fully unrolled: 20 WMMA tiles per wave-pair of tile-rows, branchless operands.
// Phase 3: anti-diagonal soft-DTW DP carrying (R, Rdot) where Rdot is the JVP
//          in direction Omega[i,j] = (i-j)^2, giving sum(E*Omega) exactly.
__global__ __launch_bounds__(160) void dilate_dp_kernel(
    const float* __restrict__ inp,   // p: (B,C,N) flattened
    const float* __restrict__ tgt,   // t: (B,C,N) flattened
    float* __restrict__ ws)          // ws[0..511]=sdtw, ws[512..1023]=loss_temporal
{
  extern __shared__ float smem[];
  float* sD  = smem;               // NN*NN   cost matrix (100 KB in 320KB WGP LDS)
  float* sT  = sD + NN * NN;       // NN
  float* sP  = sT + NN;            // NN
  float* sV  = sP + NN;            // 3*NN    rotating R diagonals
  float* sVd = sV + 3 * NN;        // 3*NN    rotating Rdot diagonals

  const int k   = blockIdx.x;      // problem index in [0, B*C)
  const int tid = threadIdx.x;

  // ---- phase 1: stage the two 160-float series ----
  sT[tid] = tgt[k * NN + tid];
  sP[tid] = inp[k * NN + tid];
  __syncthreads();

  // ---- phase 2: cost matrix via WMMA (10x10 grid of 16x16 tiles, K=4 f32) ----
  const int wave = tid >> 5;
  const int lane = tid & 31;
  const int hsel = lane >> 4;      // lane half: supplies K={0,1} or K={2,3}
  const int m    = lane & 15;      // A-row / B-col index within tile
#pragma unroll
  for (int tr = 0; tr < 2; ++tr) {
    const int ti = wave * 2 + tr;          // tile row: wave w owns rows 2w, 2w+1
    const float tv = sT[ti * 16 + m];
    v2f a;                                  // branchless: half0 -> {t^2, t}, half1 -> {1, 0}
    a[0] = hsel ? 1.0f : tv * tv;
    a[1] = hsel ? 0.0f : tv;
#pragma unroll
    for (int tj = 0; tj < 10; ++tj) {
      const float pv = sP[tj * 16 + m];
      v2f b;                                // half0 -> {1, -2p}, half1 -> {p^2, 0}
      b[0] = hsel ? pv * pv : 1.0f;
      b[1] = hsel ? 0.0f : -2.0f * pv;
      v8f c = {};
      // (neg_a, A, neg_b, B, c_mod, C, reuse_a, reuse_b)
      c = __builtin_amdgcn_wmma_f32_16x16x4_f32(false, a, false, b, (short)0, c,
                                                false, false);
      const int col  = tj * 16 + m;         // N = lane (mod 16)
      const int row0 = ti * 16 + 8 * hsel;  // VGPR r holds row M = r + 8*half
#pragma unroll
      for (int r = 0; r < 8; ++r)
        sD[(row0 + r) * NN + col] = c[r];
    }
  }
  __syncthreads();

  // ---- phase 3: wavefront DP with dual numbers (value + JVP wrt Omega) ----
  const int j   = tid;             // blockDim.x == NN: every thread owns a column
  const int jm1 = (j > 0) ? j - 1 : 0;
  float* Vw  = sV;            float* Dw  = sVd;            // diagonal d   (write)
  float* Vm1 = sV + NN;       float* Dm1 = sVd + NN;       // diagonal d-1
  float* Vm2 = sV + 2 * NN;   float* Dm2 = sVd + 2 * NN;   // diagonal d-2
  float rprev = BIGF, dprev = 0.0f;   // own value from previous diagonal (the "up" dep)

  for (int d = 0; d < 2 * NN - 1; ++d) {
    const int i = d - j;
    const bool valid = (i >= 0) && (i < NN);
    const int ic = (i < 0) ? 0 : ((i > NN - 1) ? NN - 1 : i);

    // unconditional clamped LDS loads; boundary values selected in afterwards
    const float vdiag = Vm2[jm1], gdiag = Dm2[jm1];
    const float vleft = Vm1[jm1], gleft = Dm1[jm1];
    const float dcost = sD[ic * NN + j];

    const bool top  = (i <= 0);
    const bool left = (j == 0);
    const float rd = (top && left) ? 0.0f : ((top || left) ? BIGF : vdiag);
    const float dd = (top || left) ? 0.0f : gdiag;
    const float ru = top  ? BIGF : rprev;
    const float du = top  ? 0.0f : dprev;
    const float rl = left ? BIGF : vleft;
    const float dl = left ? 0.0f : gleft;

    const float mn = fminf(rd, fminf(ru, rl));
    const float ea = __expf((mn - rd) * INV_G);   // softmin weights (min-shifted)
    const float eb = __expf((mn - ru) * INV_G);
    const float ec = __expf((mn - rl) * INV_G);
    const float S  = ea + eb + ec;                // S >= 1, always finite
    const float sm = mn - GAMMA_F * __logf(S);
    float om = (float)(i - j); om *= om;          // Omega tangent (i-j)^2

    const float rnew = dcost + sm;
    const float dnew = om + (ea * dd + eb * du + ec * dl) * __frcp_rn(S);

    if (valid) {
      Vw[j] = rnew;  Dw[j] = dnew;
      rprev = rnew;  dprev = dnew;
    }
    __syncthreads();                 // one barrier per anti-diagonal
    float* t0 = Vm2; Vm2 = Vm1; Vm1 = Vw; Vw = t0;   // rotate ring buffers
    float* t1 = Dm2; Dm2 = Dm1; Dm1 = Dw; Dw = t1;
  }

  if (tid == 0) {
    // last write (d = 2NN-2) rotated into Vm1/Dm1
    ws[k]      = Vm1[NN - 1];                                 // sdtw value
    ws[BC + k] = Dm1[NN - 1] * (1.0f / (float)(NN * NN));     // sum(E*Omega)/N^2
  }
}

// Reduce 512 per-problem results -> [loss, loss_shape(64), loss_temporal(64)]
__global__ __launch_bounds__(64) void dilate_reduce_kernel(
    const float* __restrict__ ws, float* __restrict__ out)
{
  __shared__ float red[64];
  const int b = threadIdx.x;       // 64 threads, one per batch element
  float ss = 0.0f, st = 0.0f;
#pragma unroll
  for (int c = 0; c < CCH; ++c) {
    ss += ws[b * CCH + c];
    st += ws[BC + b * CCH + c];
  }
  ss *= (1.0f / (float)CCH);
  st *= (1.0f / (float)CCH);
  out[1 + b]      = ss;            // loss_shape
  out[1 + BB + b] = st;            // loss_temporal
  red[b] = 0.5f * ss + 0.5f * st;  // ALPHA = 0.5
  __syncthreads();
#pragma unroll
  for (int s = 32; s > 0; s >>= 1) {
    if (b < s) red[b] += red[b + s];
    __syncthreads();
  }
  if (b == 0) out[0] = red[0] * (1.0f / (float)BB);
}

extern "C" void kernel_launch(void* const* d_in, const int* in_sizes, int n_in,
                              void* d_out, int out_size, void* d_ws, size_t ws_size,
                              hipStream_t stream) {
  const float* inp = (const float*)d_in[0];   // "input"  -> p
  const float* tgt = (const float*)d_in[1];   // "target" -> t
  float* out = (float*)d_out;
  float* ws  = (float*)d_ws;

  const size_t shmem = (size_t)(NN * NN + 2 * NN + 6 * NN) * sizeof(float); // 107,520 B
  dilate_dp_kernel<<<BC, 160, shmem, stream>>>(inp, tgt, ws);
  dilate_reduce_kernel<<<1, 64, 0, stream>>>(ws, out);
}